// FeatureEncoder_69595650064849
// MI455X (gfx1250) — compile-verified
//
#include <hip/hip_runtime.h>
#include <hip/hip_bf16.h>

// FeatureEncoder spike encoding for MI455X (gfx1250, wave32).
//
// Pass 1a: per-(batch, sub-block) partial min/max (64 batches x 16 sub-blocks
//          = 1024 workgroups -> full-device occupancy for the 32.8 MB read).
// Pass 1b: 64-thread combine of 16 partials per batch.
// Pass 2 : norm = 8*(x-min)/(max-min); out[..., j] = (norm > PERM[j]) ? 1 : 0.
//
// Roofline: 262 MB write-once output dominates (~11.3 us @ 23.3 TB/s), so the
// encode pass uses non-temporal global_store_b128 (th:TH_STORE_NT) to avoid
// sweeping the 192 MB L2. Input (32.8 MB) is read with B128 loads + WGP-scope
// global_prefetch_b8 in the reduction loop, and stays L2-resident for pass 2.

typedef __attribute__((ext_vector_type(4))) float v4f;

#define PER_BATCH      128000              // 1000 * 128 floats per batch
#define VEC4_PER_BATCH (PER_BATCH / 4)     // 32000
#define SUBBLOCKS      16                  // partial-reduction blocks per batch
#define VEC4_PER_SUB   (VEC4_PER_BATCH / SUBBLOCKS)  // 2000
#define P1_THREADS     256
#define P2_THREADS     256
#define TIME_STEPS     8

// ---------------------------------------------------------------------------
// Pass 1a: partial min/max. Block bi covers batch = bi/16, sub-chunk = bi%16
// (2000 contiguous float4s). 256 threads (8 wave32 waves), ~8 strided B128
// loads each, register reduce -> wave32 shfl_xor butterfly -> LDS across
// waves -> float2{min,max} partial per block.
// ---------------------------------------------------------------------------
__global__ void __launch_bounds__(P1_THREADS)
fe_partial_minmax_kernel(const float* __restrict__ x, float2* __restrict__ partial)
{
    const int batch = blockIdx.x >> 4;          // / SUBBLOCKS
    const int sub   = blockIdx.x & (SUBBLOCKS - 1);
    const v4f* xb = (const v4f*)(x) + (size_t)batch * VEC4_PER_BATCH
                                    + (size_t)sub * VEC4_PER_SUB;

    float vmin =  3.402823466e+38f;
    float vmax = -3.402823466e+38f;

    for (int i = threadIdx.x; i < VEC4_PER_SUB; i += P1_THREADS) {
        // WGP-scope prefetch of this thread's next stride (global_prefetch_b8).
        __builtin_prefetch(xb + i + P1_THREADS, 0, 3);
        const v4f v = xb[i];
        vmin = fminf(vmin, fminf(fminf(v.x, v.y), fminf(v.z, v.w)));
        vmax = fmaxf(vmax, fmaxf(fmaxf(v.x, v.y), fmaxf(v.z, v.w)));
    }

    // Wave32 butterfly reduction (warpSize == 32 on gfx1250).
    #pragma unroll
    for (int off = 16; off >= 1; off >>= 1) {
        vmin = fminf(vmin, __shfl_xor(vmin, off, 32));
        vmax = fmaxf(vmax, __shfl_xor(vmax, off, 32));
    }

    __shared__ float smin[P1_THREADS / 32];
    __shared__ float smax[P1_THREADS / 32];
    const int wave = threadIdx.x >> 5;
    const int lane = threadIdx.x & 31;
    if (lane == 0) { smin[wave] = vmin; smax[wave] = vmax; }
    __syncthreads();

    if (threadIdx.x == 0) {
        float m = smin[0], M = smax[0];
        #pragma unroll
        for (int w = 1; w < P1_THREADS / 32; ++w) {
            m = fminf(m, smin[w]);
            M = fmaxf(M, smax[w]);
        }
        partial[blockIdx.x] = make_float2(m, M);
    }
}

// ---------------------------------------------------------------------------
// Pass 1b: one thread per batch combines its 16 partials.
// ---------------------------------------------------------------------------
__global__ void fe_combine_kernel(const float2* __restrict__ partial,
                                  float2* __restrict__ mm, int batches)
{
    const int b = blockIdx.x * blockDim.x + threadIdx.x;
    if (b >= batches) return;
    float m =  3.402823466e+38f;
    float M = -3.402823466e+38f;
    #pragma unroll
    for (int s = 0; s < SUBBLOCKS; ++s) {
        const float2 p = partial[b * SUBBLOCKS + s];
        m = fminf(m, p.x);
        M = fmaxf(M, p.y);
    }
    mm[b] = make_float2(m, M);
}

// ---------------------------------------------------------------------------
// Pass 2: one float4 of input per thread -> 32 output floats (4 elements x 8
// timesteps) via 8 non-temporal B128 stores; 128 contiguous bytes per lane,
// 4 KB contiguous per wave. 32000/256 = 125 blocks per batch exactly, so the
// batch id and min/max are wave-uniform (scalar load path).
//
// Time-axis permutation baked in as compile-time thresholds; must equal
// jax.random.permutation(jax.random.key(1), 8): out[..., j] = norm > PERM[j].
// ---------------------------------------------------------------------------
__global__ void __launch_bounds__(P2_THREADS)
fe_encode_kernel(const float* __restrict__ x,
                 const float2* __restrict__ mm,
                 float* __restrict__ out,
                 int blocks_per_batch)
{
    constexpr float T[TIME_STEPS] = { 6.0f, 2.0f, 5.0f, 0.0f, 4.0f, 7.0f, 1.0f, 3.0f };

    const unsigned g = blockIdx.x * P2_THREADS + threadIdx.x;  // float4 index
    const int b = blockIdx.x / blocks_per_batch;               // uniform per block

    const float2 m     = mm[b];
    const float  mn    = m.x;
    const float  scale = (float)TIME_STEPS / (m.y - mn);

    const v4f v = ((const v4f*)x)[g];
    float* o = out + (size_t)g * (4u * TIME_STEPS);

    const float elem[4] = { v.x, v.y, v.z, v.w };
    #pragma unroll
    for (int e = 0; e < 4; ++e) {
        const float n = (elem[e] - mn) * scale;
        v4f lo, hi;
        lo.x = (n > T[0]) ? 1.0f : 0.0f;
        lo.y = (n > T[1]) ? 1.0f : 0.0f;
        lo.z = (n > T[2]) ? 1.0f : 0.0f;
        lo.w = (n > T[3]) ? 1.0f : 0.0f;
        hi.x = (n > T[4]) ? 1.0f : 0.0f;
        hi.y = (n > T[5]) ? 1.0f : 0.0f;
        hi.z = (n > T[6]) ? 1.0f : 0.0f;
        hi.w = (n > T[7]) ? 1.0f : 0.0f;
        // Write-once stream: non-temporal B128 stores, keep 262 MB out of L2.
        __builtin_nontemporal_store(lo, (v4f*)(o + e * TIME_STEPS));
        __builtin_nontemporal_store(hi, (v4f*)(o + e * TIME_STEPS + 4));
    }
}

// ---------------------------------------------------------------------------
extern "C" void kernel_launch(void* const* d_in, const int* in_sizes, int n_in,
                              void* d_out, int out_size, void* d_ws, size_t ws_size,
                              hipStream_t stream)
{
    const float* x   = (const float*)d_in[0];
    float*       out = (float*)d_out;

    const int total   = in_sizes[0];                 // 8,192,000
    const int batches = total / PER_BATCH;           // 64

    // Workspace layout: [0, batches*16) partial float2s, then batches float2s.
    float2* partial = (float2*)d_ws;                       // 1024 * 8 B = 8 KB
    float2* mm      = partial + (size_t)batches * SUBBLOCKS;

    const int p1_blocks = batches * SUBBLOCKS;             // 1024
    const int vec4_tot  = total / 4;                       // 2,048,000
    const int p2_blocks = vec4_tot / P2_THREADS;           // 8000
    const int blocks_per_batch = VEC4_PER_BATCH / P2_THREADS; // 125

    fe_partial_minmax_kernel<<<p1_blocks, P1_THREADS, 0, stream>>>(x, partial);
    fe_combine_kernel<<<1, 64, 0, stream>>>(partial, mm, batches);
    fe_encode_kernel<<<p2_blocks, P2_THREADS, 0, stream>>>(x, mm, out, blocks_per_batch);
}